// VitVQModule_78288663872164
// MI455X (gfx1250) — compile-verified
//
#include <hip/hip_runtime.h>

typedef __attribute__((ext_vector_type(2))) float v2f;
typedef __attribute__((ext_vector_type(4))) float v4f;
typedef __attribute__((ext_vector_type(8))) float v8f;

#define DIMS    32
#define NTOK    32768
#define NCODE   8192
#define VQ_EPS  1e-12f
#define CHUNK   64                    // codes staged per LDS buffer
#define NCHUNK  (NCODE / CHUNK)       // 128
#define CHUNKF  (CHUNK * DIMS)        // 2048 floats = 8192 bytes per buffer

// ---------------------------------------------------------------------------
// L2-normalize rows of a [nrows, 32] f32 matrix. One thread per row.
// ---------------------------------------------------------------------------
__global__ __launch_bounds__(256) void normalize_rows(const float* __restrict__ in,
                                                      float* __restrict__ out,
                                                      int nrows) {
    int r = blockIdx.x * blockDim.x + threadIdx.x;
    if (r >= nrows) return;
    const v4f* src = (const v4f*)(in + (size_t)r * DIMS);
    v4f v[8];
    float s = 0.f;
#pragma unroll
    for (int i = 0; i < 8; ++i) {
        v[i] = src[i];
#pragma unroll
        for (int j = 0; j < 4; ++j) s += v[i][j] * v[i][j];
    }
    float inv = rsqrtf(s + VQ_EPS);
    v4f* dst = (v4f*)(out + (size_t)r * DIMS);
#pragma unroll
    for (int i = 0; i < 8; ++i) {
        v4f o;
#pragma unroll
        for (int j = 0; j < 4; ++j) o[j] = v[i][j] * inv;
        dst[i] = o;
    }
}

// ---------------------------------------------------------------------------
// Similarity scan + argmax via V_WMMA_F32_16X16X4_F32 with async global->LDS
// double-buffered codebook staging (GLOBAL_LOAD_ASYNC_TO_LDS_B128/ASYNCcnt).
// Block = 8 waves; each wave owns a 16-token tile (A resident in VGPRs).
// The block cooperatively stages 64-code chunks into LDS; all 8 waves read
// B fragments from LDS (8x codebook-traffic reduction vs per-wave L2 reads).
// D layout: acc[v] on lane l  ->  M = v + 8*(l>=16), N = l&15.
// ---------------------------------------------------------------------------
__global__ __launch_bounds__(256) void vq_score(const float* __restrict__ zn,
                                                const float* __restrict__ en,
                                                int* __restrict__ out_idx) {
    __shared__ float smem[2 * CHUNKF];          // 2 x 8 KB double buffer

    const int tid   = threadIdx.x;
    const int lane  = tid & 31;
    const int wave  = tid >> 5;
    const int mbase = (blockIdx.x * 8 + wave) * 16;
    const int lrow  = lane & 15;
    const int koff  = (lane >> 4) * 2;          // half-wave K pair base (0 or 2)

    // A fragments: a[j] holds K = 4j+koff, 4j+koff+1 for row M = lrow
    const float* arow = zn + (size_t)(mbase + lrow) * DIMS + koff;
    v2f a[8];
#pragma unroll
    for (int j = 0; j < 8; ++j) a[j] = *(const v2f*)(arow + j * 4);

    // LDS byte offset of the staging buffer (flat LDS address truncates to
    // addr[31:0] == workgroup-relative LDS offset per CDNA5 ISA)
    const unsigned lds_base = (unsigned)(uintptr_t)(&smem[0]);

    // Stage one 8 KB chunk: 256 threads x 2 x b128 async transfers, coalesced.
#define STAGE_CHUNK(c, buf)                                                      \
    do {                                                                         \
        unsigned long long g =                                                   \
            (unsigned long long)(uintptr_t)(en + (size_t)(c) * CHUNKF) +         \
            (unsigned)(tid * 16);                                                \
        unsigned l = lds_base + (unsigned)((buf) * (CHUNKF * 4)) +               \
                     (unsigned)(tid * 16);                                       \
        asm volatile("global_load_async_to_lds_b128 %0, %1, off"                 \
                     :: "v"(l), "v"(g) : "memory");                              \
        asm volatile("global_load_async_to_lds_b128 %0, %1, off"                 \
                     :: "v"(l + 4096u), "v"(g + 4096ull) : "memory");            \
    } while (0)

    float best[8];
    int   bestn[8];
#pragma unroll
    for (int v = 0; v < 8; ++v) { best[v] = -1e30f; bestn[v] = 0; }

    STAGE_CHUNK(0, 0);
    asm volatile("s_wait_asynccnt 0x0" ::: "memory");
    __syncthreads();

    for (int c = 0; c < NCHUNK; ++c) {
        const int cur = c & 1;
        if (c + 1 < NCHUNK) STAGE_CHUNK(c + 1, cur ^ 1);   // overlap with compute

#pragma unroll
        for (int t = 0; t < 4; ++t) {                      // 4 x 16-code tiles
            const int boff = cur * CHUNKF + (t * 16 + lrow) * DIMS + koff;
            v2f b[8];
#pragma unroll
            for (int j = 0; j < 8; ++j) b[j] = *(const v2f*)(&smem[boff + j * 4]);

            v8f acc = {0.f, 0.f, 0.f, 0.f, 0.f, 0.f, 0.f, 0.f};
#pragma unroll
            for (int j = 0; j < 8; ++j)
                acc = __builtin_amdgcn_wmma_f32_16x16x4_f32(
                    false, a[j], false, b[j], (short)0, acc, false, false);

            const int myn = c * CHUNK + t * 16 + lrow;     // this lane's code idx
#pragma unroll
            for (int v = 0; v < 8; ++v) {
                // strictly-greater keeps earliest index, matching argmin ties
                if (acc[v] > best[v]) { best[v] = acc[v]; bestn[v] = myn; }
            }
        }

        if (c + 1 < NCHUNK)
            asm volatile("s_wait_asynccnt 0x0" ::: "memory");
        __syncthreads();   // all waves done with buf[cur] before it is restaged
    }
#undef STAGE_CHUNK

    // cross-lane argmax within each 16-lane half (xor masks stay inside halves)
#pragma unroll
    for (int s = 1; s < 16; s <<= 1) {
#pragma unroll
        for (int v = 0; v < 8; ++v) {
            float os = __shfl_xor(best[v], s, 32);
            int   on = __shfl_xor(bestn[v], s, 32);
            if (os > best[v] || (os == best[v] && on < bestn[v])) {
                best[v] = os; bestn[v] = on;
            }
        }
    }
    if (lane == 0) {
#pragma unroll
        for (int v = 0; v < 8; ++v) out_idx[mbase + v] = bestn[v];
    } else if (lane == 16) {
#pragma unroll
        for (int v = 0; v < 8; ++v) out_idx[mbase + 8 + v] = bestn[v];
    }
}

// ---------------------------------------------------------------------------
// Gather raw embedding by index, normalize, write z_q and index; accumulate
// per-block squared error partials (deterministic, no float atomics).
// ---------------------------------------------------------------------------
__global__ __launch_bounds__(256) void vq_output(const float* __restrict__ emb,
                                                 const float* __restrict__ zn,
                                                 const int* __restrict__ idx,
                                                 float* __restrict__ out,
                                                 float* __restrict__ partials) {
    __shared__ float red[256];
    int t = blockIdx.x * 256 + threadIdx.x;
    int k = idx[t];

    const v4f* e = (const v4f*)(emb + (size_t)k * DIMS);
    v4f v[8];
    float s = 0.f;
#pragma unroll
    for (int i = 0; i < 8; ++i) {
        v[i] = e[i];
#pragma unroll
        for (int j = 0; j < 4; ++j) s += v[i][j] * v[i][j];
    }
    float inv = rsqrtf(s + VQ_EPS);

    const v4f* h  = (const v4f*)(zn + (size_t)t * DIMS);
    v4f*       zq = (v4f*)(out + (size_t)t * DIMS);
    float err = 0.f;
#pragma unroll
    for (int i = 0; i < 8; ++i) {
        v4f hn = h[i];
        v4f o;
#pragma unroll
        for (int j = 0; j < 4; ++j) {
            o[j] = v[i][j] * inv;
            float d = o[j] - hn[j];
            err += d * d;
        }
        zq[i] = o;
    }
    out[(size_t)NTOK * DIMS + t] = (float)k;

    red[threadIdx.x] = err;
    __syncthreads();
    for (int s2 = 128; s2 > 0; s2 >>= 1) {
        if (threadIdx.x < s2) red[threadIdx.x] += red[threadIdx.x + s2];
        __syncthreads();
    }
    if (threadIdx.x == 0) partials[blockIdx.x] = red[0];
}

__global__ void vq_finalize(const float* __restrict__ partials,
                            float* __restrict__ out) {
    if (threadIdx.x == 0) {
        float s = 0.f;
        for (int i = 0; i < NTOK / 256; ++i) s += partials[i];
        float mean = s / (float)((size_t)NTOK * DIMS);
        out[(size_t)NTOK * DIMS + NTOK]     = mean;  // q_latent_loss
        out[(size_t)NTOK * DIMS + NTOK + 1] = mean;  // e_latent_loss
    }
}

extern "C" void kernel_launch(void* const* d_in, const int* in_sizes, int n_in,
                              void* d_out, int out_size, void* d_ws, size_t ws_size,
                              hipStream_t stream) {
    const float* hidden    = (const float*)d_in[0];  // [32,1024,32]
    const float* embedding = (const float*)d_in[1];  // [8192,32]
    float* out = (float*)d_out;

    // workspace layout
    float* en       = (float*)d_ws;                     // 8192*32   normalized codebook
    float* zn       = en + (size_t)NCODE * DIMS;        // 32768*32  normalized tokens
    int*   idxb     = (int*)(zn + (size_t)NTOK * DIMS); // 32768
    float* partials = (float*)(idxb + NTOK);            // 128

    normalize_rows<<<(NTOK + 255) / 256, 256, 0, stream>>>(hidden, zn, NTOK);
    normalize_rows<<<(NCODE + 255) / 256, 256, 0, stream>>>(embedding, en, NCODE);
    // 2048 token tiles / 8 waves per block = 256 blocks
    vq_score<<<NTOK / 16 / 8, 256, 0, stream>>>(zn, en, idxb);
    vq_output<<<NTOK / 256, 256, 0, stream>>>(embedding, zn, idxb, out, partials);
    vq_finalize<<<1, 32, 0, stream>>>(partials, out);
}